// SingleClusterOpsinModel_6700148982107
// MI455X (gfx1250) — compile-verified
//
#include <hip/hip_runtime.h>
#include <hip/hip_bf16.h>

typedef float v2f __attribute__((ext_vector_type(2)));
typedef float v8f __attribute__((ext_vector_type(8)));

#define LCH 256           // chunk length (elements)
#define PAD 256           // zero front-padding for shifted-U gathers
#define NC  8192          // T / LCH, T = 2097152
#define GRP 128           // chunks per scan group
#define NG  (NC / GRP)    // 64 groups

// workspace layout (float offsets)
#define OFF_H    0                        // h[k] = c^T A^k b,         256
#define OFF_R    256                      // R[i][8] = c^T A^{i+1},    256*8
#define OFF_P    (OFF_R + 2048)           // P[i][8] = A^{L-1-i} b,    256*8
#define OFF_M    (OFF_P + 2048)           // M = A^256                 64
#define OFF_M128 (OFF_M + 64)             // M^128                     64
#define OFF_MPOW (OFF_M128 + 64)          // M^k, k=0..127             128*64
#define OFF_V    (OFF_MPOW + 128*64)      // per-chunk v_j             NC*8
#define OFF_XL   (OFF_V + NC*8)           // per-chunk local state     NC*8
#define OFF_E    (OFF_XL + NC*8)          // per-group end state       NG*8
#define OFF_XG   (OFF_E + NG*8)           // per-group input state     NG*8
// total = OFF_XG + NG*8 floats  (~580 KB)

// ---------------- Kernel 1: tiny precompute (one block, 64 threads) ----------
__global__ void k_precompute(const float* __restrict__ A,
                             const float* __restrict__ Bo,
                             const float* __restrict__ Co,
                             float* __restrict__ ws)
{
    __shared__ float sA[64], sr[8], sp[8], sb[8], t0[64], t1[64];
    const int tid = threadIdx.x;           // 64 threads
    sA[tid] = A[tid];
    if (tid < 8) { sb[tid] = Bo[tid]; sr[tid] = Co[tid]; sp[tid] = Bo[tid]; }
    __syncthreads();

    float* h = ws + OFF_H;
    float* R = ws + OFF_R;
    float* P = ws + OFF_P;

    // h[k] = r_k . b ; R[k] = r_{k+1} = c^T A^{k+1} ; P[255-k] = p_k = A^k b
    for (int k = 0; k < LCH; ++k) {
        float nr = 0.f, np = 0.f;
        if (tid == 0) {
            float hv = 0.f;
            for (int m = 0; m < 8; ++m) hv += sr[m] * sb[m];
            h[k] = hv;
        }
        if (tid < 8) {
            for (int m = 0; m < 8; ++m) {
                nr += sr[m] * sA[m * 8 + tid];      // r <- r A
                np += sA[tid * 8 + m] * sp[m];      // p <- A p
            }
        }
        __syncthreads();
        if (tid < 8) {
            P[(LCH - 1 - k) * 8 + tid] = sp[tid];   // old p = A^k b
            sr[tid] = nr; sp[tid] = np;
        }
        __syncthreads();
        if (tid < 8) R[k * 8 + tid] = sr[tid];      // c^T A^{k+1}
        __syncthreads();
    }

    // M = A^256 by 8 squarings (thread per matrix element)
    const int rr = tid >> 3, cc = tid & 7;
    t0[tid] = sA[tid];
    __syncthreads();
    for (int s = 0; s < 8; ++s) {
        float acc = 0.f;
        for (int m = 0; m < 8; ++m) acc += t0[rr * 8 + m] * t0[m * 8 + cc];
        __syncthreads();
        t0[tid] = acc;
        __syncthreads();
    }
    ws[OFF_M + tid] = t0[tid];

    // Mpow[k] = M^k, k = 0..127 ; M128 = M^128
    t1[tid] = (rr == cc) ? 1.f : 0.f;
    __syncthreads();
    ws[OFF_MPOW + tid] = t1[tid];
    for (int k = 0; k < GRP - 1; ++k) {
        float acc = 0.f;
        for (int m = 0; m < 8; ++m) acc += t1[rr * 8 + m] * t0[m * 8 + cc];
        __syncthreads();
        t1[tid] = acc;
        __syncthreads();
        ws[OFF_MPOW + (k + 1) * 64 + tid] = acc;
    }
    {
        float acc = 0.f;
        for (int m = 0; m < 8; ++m) acc += t1[rr * 8 + m] * t0[m * 8 + cc];
        ws[OFF_M128 + tid] = acc;
    }
}

// ---------------- Kernel 2: per-chunk carry vectors v_j ----------------------
__global__ __launch_bounds__(256) void k_chunkv(const float* __restrict__ u,
                                                const float* __restrict__ betap,
                                                float* __restrict__ ws)
{
    const int w = threadIdx.x >> 5, lane = threadIdx.x & 31;
    const int chunk = blockIdx.x * 8 + w;
    const float beta = betap[0];
    const float* P = ws + OFF_P;
    const long base = (long)chunk * LCH;

    float a0=0,a1=0,a2=0,a3=0,a4=0,a5=0,a6=0,a7=0;
    for (int t = 0; t < 8; ++t) {
        const int i = t * 32 + lane;
        const float uv = beta * u[base + i];
        const float4 p0 = ((const float4*)(P + i * 8))[0];
        const float4 p1 = ((const float4*)(P + i * 8))[1];
        a0 += p0.x * uv; a1 += p0.y * uv; a2 += p0.z * uv; a3 += p0.w * uv;
        a4 += p1.x * uv; a5 += p1.y * uv; a6 += p1.z * uv; a7 += p1.w * uv;
    }
    for (int off = 16; off >= 1; off >>= 1) {
        a0 += __shfl_xor(a0, off, 32); a1 += __shfl_xor(a1, off, 32);
        a2 += __shfl_xor(a2, off, 32); a3 += __shfl_xor(a3, off, 32);
        a4 += __shfl_xor(a4, off, 32); a5 += __shfl_xor(a5, off, 32);
        a6 += __shfl_xor(a6, off, 32); a7 += __shfl_xor(a7, off, 32);
    }
    if (lane == 0) {
        float* v = ws + OFF_V + (long)chunk * 8;
        ((float4*)v)[0] = make_float4(a0, a1, a2, a3);
        ((float4*)v)[1] = make_float4(a4, a5, a6, a7);
    }
}

// ---------------- Kernel 3a: parallel group scans (64 blocks x 1 wave) -------
__global__ void k_scan_groups(float* __restrict__ ws)
{
    const int g = blockIdx.x;
    const int lane = threadIdx.x;            // 32
    const int r = (lane < 8) ? lane : 0;
    float m[8];
    #pragma unroll
    for (int c = 0; c < 8; ++c) m[c] = ws[OFF_M + r * 8 + c];

    const float* v = ws + OFF_V;
    float* xl = ws + OFF_XL;
    float x = 0.f;
    for (int l = 0; l < GRP; ++l) {
        const int j = g * GRP + l;
        if (lane < 8) xl[j * 8 + lane] = x;           // state entering chunk j
        float nx = 0.f;
        #pragma unroll
        for (int c = 0; c < 8; ++c) nx += m[c] * __shfl(x, c, 32);
        nx += v[j * 8 + r];                           // only lanes<8 are read back
        x = nx;
    }
    if (lane < 8) ws[OFF_E + g * 8 + lane] = x;
}

// ---------------- Kernel 3b: top-level scan over 64 groups (1 wave) ----------
__global__ void k_scan_top(float* __restrict__ ws)
{
    const int lane = threadIdx.x;            // 32
    const int r = (lane < 8) ? lane : 0;
    float m[8];
    #pragma unroll
    for (int c = 0; c < 8; ++c) m[c] = ws[OFF_M128 + r * 8 + c];

    const float* e = ws + OFF_E;
    float* Xg = ws + OFF_XG;
    float x = 0.f;
    if (lane < 8) Xg[lane] = 0.f;
    for (int g = 0; g < NG - 1; ++g) {
        float nx = 0.f;
        #pragma unroll
        for (int c = 0; c < 8; ++c) nx += m[c] * __shfl(x, c, 32);
        nx += e[g * 8 + r];
        x = nx;
        if (lane < 8) Xg[(g + 1) * 8 + lane] = x;
    }
}

// ---------------- Kernel 4: main — WMMA triangular-Toeplitz conv + cross -----
__global__ __launch_bounds__(256) void k_main(const float* __restrict__ u,
                                              const float* __restrict__ betap,
                                              const float* __restrict__ ws,
                                              float* __restrict__ out)
{
    // hr = reversed impulse response: hr[q] = h[255-q], hr[256..271] = 0.
    // A fragment {h[ia], h[ia-1]} == {hr[255-ia], hr[256-ia]} -> one ascending
    // paired LDS load, padding supplies the ia<=0 zeros (triangular mask).
    // us is front-padded with PAD zeros per wave so the column-shifted U gather
    // (n0-d)*16+col is always in-range; the n0<d zeros come from the padding.
    __shared__ float hr[LCH + 16];
    __shared__ float Rs[LCH * 8];
    __shared__ float us[8][PAD + LCH];
    __shared__ float ys[8][LCH];

    const int tid = threadIdx.x;
    const int w = tid >> 5, lane = tid & 31;
    const int n0 = lane & 15, half = lane >> 4;
    const float beta = betap[0];

    // cooperative LDS fills
    hr[tid] = ws[OFF_H + (LCH - 1) - tid];
    if (tid < 16) hr[LCH + tid] = 0.f;
    {
        const float4* Rg = (const float4*)(ws + OFF_R);
        float4* Rl = (float4*)Rs;
        Rl[tid * 2]     = Rg[tid * 2];
        Rl[tid * 2 + 1] = Rg[tid * 2 + 1];
    }
    const int j = blockIdx.x * 8 + w;        // chunk handled by this wave
    const long base = (long)j * LCH;
    {
        float4* zp = (float4*)us[w];                  // zero the pad
        zp[lane * 2]     = make_float4(0.f, 0.f, 0.f, 0.f);
        zp[lane * 2 + 1] = make_float4(0.f, 0.f, 0.f, 0.f);
        const float4* ug = (const float4*)(u + base);
        float4 u0 = ug[lane * 2], u1 = ug[lane * 2 + 1];
        u0.x *= beta; u0.y *= beta; u0.z *= beta; u0.w *= beta;
        u1.x *= beta; u1.y *= beta; u1.z *= beta; u1.w *= beta;
        float4* ul = (float4*)(us[w] + PAD);
        ul[lane * 2] = u0; ul[lane * 2 + 1] = u1;
    }
    __syncthreads();

    // reconstruct chunk input state: x_in = Mpow[l] @ Xg[g] + xlocal[j]
    const int g = j >> 7, l = j & 127;
    const int r = (lane < 8) ? lane : 0;
    float xr = ws[OFF_XL + (long)j * 8 + r];
    {
        const float* mp = ws + OFF_MPOW + l * 64 + r * 8;
        const float* xg = ws + OFF_XG + g * 8;
        #pragma unroll
        for (int c = 0; c < 8; ++c) xr += mp[c] * xg[c];
    }
    float xin[8];
    #pragma unroll
    for (int c = 0; c < 8; ++c) xin[c] = __shfl(xr, c, 32);

    // cross term: D[row][col] = R[col*16+row] . x_in   (C/D layout: vgpr rr ->
    // row rr (lanes 0-15) / rr+8 (lanes 16-31), col = n0)
    v8f D;
    #pragma unroll
    for (int rr = 0; rr < 8; ++rr) {
        const int row = rr + half * 8;
        const float* Ri = &Rs[(n0 * 16 + row) * 8];
        float acc = 0.f;
        #pragma unroll
        for (int c = 0; c < 8; ++c) acc += Ri[c] * xin[c];
        D[rr] = acc;
    }

    // local causal conv: Y += sum_d H_d(16x16) @ ColShift_d(U), H_d[r][c] =
    // h[16d + r - c] (zero-padded for the d = 0 triangle).
    // Each block-matmul = 4 x V_WMMA_F32_16X16X4_F32.
    // A frag (16x4 f32): lane<16 -> row n0, K={0,1}; lane>=16 -> row n0, K={2,3}.
    // B frag (4x16 f32): vgpr0 -> K=0 (lanes<16) / K=2 ; vgpr1 -> K=1 / K=3.
    const float* uwp = us[w] + PAD;          // padded base: gathers never masked
    #pragma unroll 4
    for (int d = 0; d < 16; ++d) {
        const int mrow = (n0 - d) * 16;      // may be negative: lands in pad
        #pragma unroll
        for (int kk = 0; kk < 4; ++kk) {
            const int col = 4 * kk + 2 * half;       // H column / U row index
            const int ia = 16 * d + n0 - col;        // >= -15 always
            v2f a, b;
            const float* ap = &hr[(LCH - 1) - ia];   // {h[ia], h[ia-1]} ascending
            a.x = ap[0];
            a.y = ap[1];
            const float* up = uwp + mrow + col;      // 8B-aligned (col even)
            b.x = up[0];
            b.y = up[1];
            D = __builtin_amdgcn_wmma_f32_16x16x4_f32(
                    false, a, false, b, (short)0, D, false, false);
        }
    }

    // transpose through LDS for coalesced b128 stores
    #pragma unroll
    for (int rr = 0; rr < 8; ++rr)
        ys[w][n0 * 16 + rr + half * 8] = D[rr];
    __syncthreads();
    {
        const float4* yl = (const float4*)ys[w];
        float4 o0 = yl[lane * 2], o1 = yl[lane * 2 + 1];
        float4* og = (float4*)(out + base);
        og[lane * 2] = o0; og[lane * 2 + 1] = o1;
    }
}

// ---------------- Launch -----------------------------------------------------
extern "C" void kernel_launch(void* const* d_in, const int* in_sizes, int n_in,
                              void* d_out, int out_size, void* d_ws, size_t ws_size,
                              hipStream_t stream)
{
    const float* u    = (const float*)d_in[0];   // (T,)
    const float* A    = (const float*)d_in[1];   // (8,8)
    const float* Bo   = (const float*)d_in[2];   // (8,1)
    const float* Co   = (const float*)d_in[3];   // (1,8)
    const float* beta = (const float*)d_in[4];   // scalar
    float* ws  = (float*)d_ws;
    float* out = (float*)d_out;

    k_precompute<<<1, 64, 0, stream>>>(A, Bo, Co, ws);
    k_chunkv<<<NC / 8, 256, 0, stream>>>(u, beta, ws);
    k_scan_groups<<<NG, 32, 0, stream>>>(ws);
    k_scan_top<<<1, 32, 0, stream>>>(ws);
    k_main<<<NC / 8, 256, 0, stream>>>(u, beta, ws, out);
}